// NeuralThickeningNet_12919261627020
// MI455X (gfx1250) — compile-verified
//
#include <hip/hip_runtime.h>
#include <hip/hip_bf16.h>
#include <math.h>
#include <stdint.h>

#define NF 100000
#define NV 50000
#define LEAKY_SLOPE 0.2f
#define LN_EPS 1e-5f
#define NRM_EPS 1e-12f

typedef __attribute__((ext_vector_type(2))) float v2f;
typedef __attribute__((ext_vector_type(8))) float v8f;

// async global -> LDS, 16 bytes per lane (ASYNCcnt-tracked, bypasses VGPRs)
__device__ __forceinline__ void async_g2l_b128(uint32_t lds_off,
                                               const void* base,
                                               uint32_t byte_off) {
  asm volatile("global_load_async_to_lds_b128 %0, %1, %2"
               :
               : "v"(lds_off), "v"(byte_off), "s"(base)
               : "memory");
}
__device__ __forceinline__ void wait_asynccnt0() {
  asm volatile("s_wait_asynccnt 0x0" ::: "memory");
}

// ---------------------------------------------------------------- utilities
__global__ void zero_kernel(float* __restrict__ p, int n) {
  int i = blockIdx.x * blockDim.x + threadIdx.x;
  if (i < n) p[i] = 0.f;
}

// ------------------------------------------------- face geometry + scatter
__global__ void geom_kernel(const float* __restrict__ verts,
                            const int* __restrict__ faces,
                            float* __restrict__ fnorm,
                            float* __restrict__ feat0,
                            float* __restrict__ vnacc,
                            float* __restrict__ faces_out) {
  int f = blockIdx.x * blockDim.x + threadIdx.x;
  if (f >= NF) return;
  int i0 = faces[3 * f], i1 = faces[3 * f + 1], i2 = faces[3 * f + 2];
  float ax = verts[3 * i0], ay = verts[3 * i0 + 1], az = verts[3 * i0 + 2];
  float bx = verts[3 * i1], by = verts[3 * i1 + 1], bz = verts[3 * i1 + 2];
  float cx = verts[3 * i2], cy = verts[3 * i2 + 1], cz = verts[3 * i2 + 2];
  float e1x = bx - ax, e1y = by - ay, e1z = bz - az;
  float e2x = cx - ax, e2y = cy - ay, e2z = cz - az;
  float crx = e1y * e2z - e1z * e2y;
  float cry = e1z * e2x - e1x * e2z;
  float crz = e1x * e2y - e1y * e2x;
  // face normal = normalize(cross + 1e-12)
  float nx = crx + 1e-12f, ny = cry + 1e-12f, nz = crz + 1e-12f;
  float nn = sqrtf(nx * nx + ny * ny + nz * nz);
  float inv = 1.f / fmaxf(nn, NRM_EPS);
  nx *= inv; ny *= inv; nz *= inv;
  fnorm[3 * f] = nx; fnorm[3 * f + 1] = ny; fnorm[3 * f + 2] = nz;
  float area = 0.5f * sqrtf(crx * crx + cry * cry + crz * crz);
  const float third = 1.f / 3.f;
  feat0[8 * f + 0] = (ax + bx + cx) * third;
  feat0[8 * f + 1] = (ay + by + cy) * third;
  feat0[8 * f + 2] = (az + bz + cz) * third;
  feat0[8 * f + 3] = nx;
  feat0[8 * f + 4] = ny;
  feat0[8 * f + 5] = nz;
  feat0[8 * f + 6] = area;
  atomicAdd(&vnacc[3 * i0 + 0], crx); atomicAdd(&vnacc[3 * i0 + 1], cry); atomicAdd(&vnacc[3 * i0 + 2], crz);
  atomicAdd(&vnacc[3 * i1 + 0], crx); atomicAdd(&vnacc[3 * i1 + 1], cry); atomicAdd(&vnacc[3 * i1 + 2], crz);
  atomicAdd(&vnacc[3 * i2 + 0], crx); atomicAdd(&vnacc[3 * i2 + 1], cry); atomicAdd(&vnacc[3 * i2 + 2], crz);
  faces_out[3 * f + 0] = (float)i0;
  faces_out[3 * f + 1] = (float)i1;
  faces_out[3 * f + 2] = (float)i2;
}

__global__ void avgdot_kernel(const float* __restrict__ fnorm,
                              const int* __restrict__ adj,
                              float* __restrict__ feat0) {
  int f = blockIdx.x * blockDim.x + threadIdx.x;
  if (f >= NF) return;
  float nx = fnorm[3 * f], ny = fnorm[3 * f + 1], nz = fnorm[3 * f + 2];
  float s = 0.f, cnt = 0.f;
#pragma unroll
  for (int j = 0; j < 3; ++j) {
    int a = adj[3 * f + j];
    if (a >= 0) {
      s += fnorm[3 * a] * nx + fnorm[3 * a + 1] * ny + fnorm[3 * a + 2] * nz;
      cnt += 1.f;
    }
  }
  feat0[8 * f + 7] = 1.f - s / fmaxf(cnt, 1.f);
}

__global__ void vertex_out_kernel(const float* __restrict__ verts,
                                  const float* __restrict__ vnacc,
                                  float* __restrict__ out_v,
                                  float* __restrict__ out_vn) {
  int v = blockIdx.x * blockDim.x + threadIdx.x;
  if (v >= NV) return;
  out_v[3 * v + 0] = verts[3 * v + 0];
  out_v[3 * v + 1] = verts[3 * v + 1];
  out_v[3 * v + 2] = verts[3 * v + 2];
  float x = vnacc[3 * v], y = vnacc[3 * v + 1], z = vnacc[3 * v + 2];
  float n = sqrtf(x * x + y * y + z * z);
  float inv = 1.f / fmaxf(n, NRM_EPS);
  out_vn[3 * v + 0] = x * inv;
  out_vn[3 * v + 1] = y * inv;
  out_vn[3 * v + 2] = z * inv;
}

// --------------------------------------------- masked neighbor mean gather
__global__ void nb_gather_kernel(const float* __restrict__ feat,
                                 const int* __restrict__ adj,
                                 float* __restrict__ nb, int in_dim) {
  int f = blockIdx.x;
  int c = threadIdx.x;
  if (c >= in_dim) return;
  int a0 = adj[3 * f], a1 = adj[3 * f + 1], a2 = adj[3 * f + 2];
  float s = 0.f, cnt = 0.f;
  if (a0 >= 0) { s += feat[(size_t)a0 * in_dim + c]; cnt += 1.f; }
  if (a1 >= 0) { s += feat[(size_t)a1 * in_dim + c]; cnt += 1.f; }
  if (a2 >= 0) { s += feat[(size_t)a2 * in_dim + c]; cnt += 1.f; }
  nb[(size_t)f * in_dim + c] = s / fmaxf(cnt, 1.f);
}

// --------------------------------------------------- global context (mean)
__global__ void colsum_kernel(const float* __restrict__ feat,
                              float* __restrict__ gsum, int M) {
  int c = threadIdx.x;                 // 256 columns
  int f0 = blockIdx.x * 1024;
  int f1 = f0 + 1024; if (f1 > M) f1 = M;
  float s = 0.f;
  for (int f = f0; f < f1; ++f) s += feat[(size_t)f * 256 + c];
  atomicAdd(&gsum[c], s);
}

__global__ void gctx_kernel(const float* __restrict__ gsum,
                            float* __restrict__ gctx, float invM) {
  int c = threadIdx.x;
  gctx[c] = gsum[c] * invM;
}

// ------------------------------------------------------------- WMMA GEMM
// Y[M,N] = act( X[M,K] @ W[N,K]^T + bias ),  X formed on the fly:
//   xmode 0: X = X0 (K = K0)
//   xmode 1: X = concat(X0[M,K0], X1[M,K0])          (K = 2*K0)
//   xmode 2: X = concat(X0[M,K0], broadcast X1[K0])  (K = 2*K0)
// actmode 0: LayerNorm(gamma,beta) + leaky;  actmode 1: leaky only
// Block: 256 thr = 8 waves = 4 (M) x 2 (N).  64 rows/block, each wave 16xN/2.
template <int NT>   // total 16-wide N tiles: N = NT*16; per-wave NTW = NT/2
__global__ __launch_bounds__(256) void gemm_act_kernel(
    const float* __restrict__ X0, const float* __restrict__ X1, int xmode,
    int K0, const float* __restrict__ W, const float* __restrict__ bias,
    const float* __restrict__ gamma, const float* __restrict__ beta,
    float* __restrict__ Y, int M, int actmode) {
  constexpr int N = NT * 16;
  constexpr int NTW = NT / 2;
  const int K = (xmode != 0) ? 2 * K0 : K0;

  __shared__ float sX[64 * 36];    // 64 rows x 32 k, pad 36 (16B-aligned slots)
  __shared__ float sW[N * 36];     // N rows (n-major) x 32 k, pad 36
  __shared__ float sS[64 * 2];     // LN cross-wave partial sums
  __shared__ float sQ[64 * 2];

  const int t = threadIdx.x;
  const int lane = t & 31;
  const int wave = t >> 5;
  const int mw = wave >> 1;              // 0..3 : row group
  const int nw = wave & 1;               // 0..1 : column half
  const int l15 = lane & 15;
  const int lhalf = lane >> 4;
  const int mbase = blockIdx.x * 64;
  const int bbase = nw * (N / 2);

  // staging maps
  const int kk_ld = t & 31;              // scalar path: k within chunk
  const int grp8 = t >> 5;               // scalar path: row sub-group
  const uint32_t kk4 = (uint32_t)(t & 7) * 4u;  // async path: k (x4 floats)
  const uint32_t rr = (uint32_t)(t >> 3);       // async path: 0..31 row

  v8f acc[NTW];
#pragma unroll
  for (int nt = 0; nt < NTW; ++nt)
#pragma unroll
    for (int r = 0; r < 8; ++r) acc[nt][r] = 0.f;

  const bool mfull = (mbase + 64 <= M);

  for (int k0 = 0; k0 < K; k0 += 32) {
    // ---- stage X tile
    const float* xb = nullptr;
    int koff = 0;
    if (mfull && k0 + 32 <= K) {
      if (k0 + 32 <= K0) { xb = X0; koff = k0; }
      else if (k0 >= K0 && xmode == 1) { xb = X1; koff = k0 - K0; }
    }
    if (xb) {  // interior: async global->LDS, 16B per lane
#pragma unroll
      for (int rb = 0; rb < 64; rb += 32) {
        uint32_t r = (uint32_t)rb + rr;
        uint32_t goff = (((uint32_t)mbase + r) * (uint32_t)K0 +
                         (uint32_t)koff + kk4) * 4u;
        async_g2l_b128((uint32_t)(uintptr_t)&sX[r * 36 + kk4], xb, goff);
      }
    } else {   // edge / concat-vector / tiny-K: zero-padded scalar staging
#pragma unroll
      for (int rb = 0; rb < 64; rb += 8) {
        int r = rb + grp8;
        int m = mbase + r;
        int k = k0 + kk_ld;
        float v = 0.f;
        if (m < M && k < K) {
          if (k < K0)          v = X0[(size_t)m * K0 + k];
          else if (xmode == 1) v = X1[(size_t)m * K0 + (k - K0)];
          else                 v = X1[k - K0];
        }
        sX[r * 36 + kk_ld] = v;
      }
    }
    // ---- stage W tile (n-major)
    if (k0 + 32 <= K) {
#pragma unroll
      for (int nb = 0; nb < N; nb += 32) {
        uint32_t n = (uint32_t)nb + rr;
        uint32_t goff = (n * (uint32_t)K + (uint32_t)k0 + kk4) * 4u;
        async_g2l_b128((uint32_t)(uintptr_t)&sW[n * 36 + kk4], W, goff);
      }
    } else {
#pragma unroll
      for (int nb = 0; nb < N; nb += 8) {
        int n = nb + grp8;
        int k = k0 + kk_ld;
        sW[n * 36 + kk_ld] = (k < K) ? W[(size_t)n * K + k] : 0.f;
      }
    }
    wait_asynccnt0();
    __syncthreads();

    // ---- 16x16x4 f32 WMMA over the 32-wide K chunk
    const int arow = (mw * 16 + l15) * 36;
#pragma unroll
    for (int kk = 0; kk < 32; kk += 4) {
      // A frag: lanes 0-15 K=kk,kk+1 ; lanes 16-31 K=kk+2,kk+3
      v2f a = *(const v2f*)&sX[arow + kk + 2 * lhalf];
#pragma unroll
      for (int nt = 0; nt < NTW; ++nt) {
        v2f b = *(const v2f*)&sW[(bbase + nt * 16 + l15) * 36 + kk + 2 * lhalf];
        acc[nt] = __builtin_amdgcn_wmma_f32_16x16x4_f32(
            false, a, false, b, (short)0, acc[nt], false, false);
      }
    }
    __syncthreads();
  }

  // ---- epilogue: bias
#pragma unroll
  for (int nt = 0; nt < NTW; ++nt) {
    float bv = bias[bbase + nt * 16 + l15];
#pragma unroll
    for (int r = 0; r < 8; ++r) acc[nt][r] += bv;
  }

  if (actmode == 0) {
    // LayerNorm over full N: local half-wave reduce, then cross-wave via LDS
    float s[8], q[8];
#pragma unroll
    for (int r = 0; r < 8; ++r) { s[r] = 0.f; q[r] = 0.f; }
#pragma unroll
    for (int nt = 0; nt < NTW; ++nt)
#pragma unroll
      for (int r = 0; r < 8; ++r) {
        float x = acc[nt][r];
        s[r] += x; q[r] += x * x;
      }
#pragma unroll
    for (int mask = 1; mask <= 8; mask <<= 1)
#pragma unroll
      for (int r = 0; r < 8; ++r) {
        s[r] += __shfl_xor(s[r], mask, 32);
        q[r] += __shfl_xor(q[r], mask, 32);
      }
    if (l15 == 0) {
#pragma unroll
      for (int r = 0; r < 8; ++r) {
        int row = mw * 16 + 8 * lhalf + r;
        sS[row * 2 + nw] = s[r];
        sQ[row * 2 + nw] = q[r];
      }
    }
    __syncthreads();
    float mu[8], is[8];
    const float invN = 1.f / (float)N;
#pragma unroll
    for (int r = 0; r < 8; ++r) {
      int row = mw * 16 + 8 * lhalf + r;
      float st = sS[row * 2 + 0] + sS[row * 2 + 1];
      float qt = sQ[row * 2 + 0] + sQ[row * 2 + 1];
      float m = st * invN;
      float v = qt * invN - m * m;
      mu[r] = m;
      is[r] = rsqrtf(v + LN_EPS);
    }
#pragma unroll
    for (int nt = 0; nt < NTW; ++nt) {
      float g = gamma[bbase + nt * 16 + l15];
      float be = beta[bbase + nt * 16 + l15];
#pragma unroll
      for (int r = 0; r < 8; ++r) {
        float x = (acc[nt][r] - mu[r]) * is[r] * g + be;
        acc[nt][r] = (x > 0.f) ? x : LEAKY_SLOPE * x;
      }
    }
  } else {
#pragma unroll
    for (int nt = 0; nt < NTW; ++nt)
#pragma unroll
      for (int r = 0; r < 8; ++r) {
        float x = acc[nt][r];
        acc[nt][r] = (x > 0.f) ? x : LEAKY_SLOPE * x;
      }
  }

  // ---- store (C layout: VGPR r -> row r + 8*lhalf, col = tile*16 + l15)
#pragma unroll
  for (int nt = 0; nt < NTW; ++nt)
#pragma unroll
    for (int r = 0; r < 8; ++r) {
      int m = mbase + mw * 16 + 8 * lhalf + r;
      if (m < M) Y[(size_t)m * N + bbase + nt * 16 + l15] = acc[nt][r];
    }
}

// ---------------------------------------------- head final: 128 -> sigmoid
__global__ void head_final_kernel(const float* __restrict__ H,
                                  const float* __restrict__ Wv,
                                  const float* __restrict__ b,
                                  const float* __restrict__ scale_ptr,
                                  float* __restrict__ out, int M) {
  int gwave = (int)((blockIdx.x * blockDim.x + threadIdx.x) >> 5);
  int lane = threadIdx.x & 31;
  if (gwave >= M) return;
  float s = 0.f;
#pragma unroll
  for (int j = 0; j < 4; ++j) {
    int k = lane + 32 * j;
    s += H[(size_t)gwave * 128 + k] * Wv[k];
  }
#pragma unroll
  for (int mask = 1; mask <= 16; mask <<= 1) s += __shfl_xor(s, mask, 32);
  if (lane == 0) {
    float z = s + b[0];
    float p = 1.f / (1.f + expf(-z));
    if (scale_ptr) p *= scale_ptr[0];
    out[gwave] = p;
  }
}

// ======================================================================
extern "C" void kernel_launch(void* const* d_in, const int* in_sizes, int n_in,
                              void* d_out, int out_size, void* d_ws,
                              size_t ws_size, hipStream_t stream) {
  (void)in_sizes; (void)n_in; (void)out_size; (void)ws_size;
  const float* vertices = (const float*)d_in[0];
  const int*   faces    = (const int*)d_in[1];
  const int*   adj      = (const int*)d_in[2];
  const float* thick    = (const float*)d_in[3];
  // params flattened in jax pytree (sorted-key) order:
  // cls[0..2]{W,b}, gnn0{W,b,beta,g}, gnn1, gnn2, reg[0..2]{W,b}
  const float* cls0W = (const float*)d_in[4];
  const float* cls0b = (const float*)d_in[5];
  const float* cls1W = (const float*)d_in[6];
  const float* cls1b = (const float*)d_in[7];
  const float* cls2W = (const float*)d_in[8];
  const float* cls2b = (const float*)d_in[9];
  const float* gW[3], *gb[3], *gbeta[3], *gg[3];
  for (int i = 0; i < 3; ++i) {
    gW[i]    = (const float*)d_in[10 + 4 * i + 0];
    gb[i]    = (const float*)d_in[10 + 4 * i + 1];
    gbeta[i] = (const float*)d_in[10 + 4 * i + 2];
    gg[i]    = (const float*)d_in[10 + 4 * i + 3];
  }
  const float* reg0W = (const float*)d_in[22];
  const float* reg0b = (const float*)d_in[23];
  const float* reg1W = (const float*)d_in[24];
  const float* reg1b = (const float*)d_in[25];
  const float* reg2W = (const float*)d_in[26];
  const float* reg2b = (const float*)d_in[27];

  // output layout (floats): verts | faces | probs | mags | vertex_normals
  float* out_v  = (float*)d_out;
  float* out_f  = out_v + 3 * NV;                 // 150000
  float* out_p  = out_f + 3 * NF;                 // 450000
  float* out_m  = out_p + NF;                     // 550000
  float* out_vn = out_m + NF;                     // 650000

  // workspace layout
  float* w = (float*)d_ws;
  float* fnorm = w;  w += 3 * (size_t)NF;
  float* feat0 = w;  w += 8 * (size_t)NF;
  float* featA = w;  w += 256 * (size_t)NF;
  float* featB = w;  w += 256 * (size_t)NF;
  float* nbbuf = w;  w += 256 * (size_t)NF;       // neighbor-mean / H2 scratch
  float* vnacc = w;  w += 3 * (size_t)NV;
  float* gsum  = w;  w += 256;
  float* gctx  = w;  w += 256;

  const int TB = 256;
  const int gF = (NF + TB - 1) / TB;
  const int gV = (NV + TB - 1) / TB;
  const int gGemm = (NF + 63) / 64;

  // 0) zero accumulators (vnacc + gsum are contiguous)
  zero_kernel<<<(3 * NV + 256 + TB - 1) / TB, TB, 0, stream>>>(vnacc, 3 * NV + 256);

  // 1) geometry, face normals, vertex-normal scatter, faces passthrough
  geom_kernel<<<gF, TB, 0, stream>>>(vertices, faces, fnorm, feat0, vnacc, out_f);
  // 2) neighbor-dot feature
  avgdot_kernel<<<gF, TB, 0, stream>>>(fnorm, adj, feat0);
  // 3) vertex outputs (vertices copy + normalized vertex normals)
  vertex_out_kernel<<<gV, TB, 0, stream>>>(vertices, vnacc, out_v, out_vn);

  // 4) GNN layer 0: K0=8 -> K=16, N=256
  nb_gather_kernel<<<NF, 32, 0, stream>>>(feat0, adj, nbbuf, 8);
  gemm_act_kernel<16><<<gGemm, TB, 0, stream>>>(feat0, nbbuf, 1, 8, gW[0], gb[0],
                                                gg[0], gbeta[0], featA, NF, 0);
  // 5) GNN layer 1
  nb_gather_kernel<<<NF, 256, 0, stream>>>(featA, adj, nbbuf, 256);
  gemm_act_kernel<16><<<gGemm, TB, 0, stream>>>(featA, nbbuf, 1, 256, gW[1], gb[1],
                                                gg[1], gbeta[1], featB, NF, 0);
  // 6) GNN layer 2
  nb_gather_kernel<<<NF, 256, 0, stream>>>(featB, adj, nbbuf, 256);
  gemm_act_kernel<16><<<gGemm, TB, 0, stream>>>(featB, nbbuf, 1, 256, gW[2], gb[2],
                                                gg[2], gbeta[2], featA, NF, 0);

  // 7) global context
  colsum_kernel<<<(NF + 1023) / 1024, 256, 0, stream>>>(featA, gsum, NF);
  gctx_kernel<<<1, 256, 0, stream>>>(gsum, gctx, 1.f / (float)NF);

  // 8) classification head
  gemm_act_kernel<16><<<gGemm, TB, 0, stream>>>(featA, gctx, 2, 256, cls0W, cls0b,
                                                nullptr, nullptr, featB, NF, 1);
  gemm_act_kernel<8><<<gGemm, TB, 0, stream>>>(featB, nullptr, 0, 256, cls1W, cls1b,
                                               nullptr, nullptr, nbbuf, NF, 1);
  head_final_kernel<<<(NF + 7) / 8, TB, 0, stream>>>(nbbuf, cls2W, cls2b, nullptr,
                                                     out_p, NF);

  // 9) regression head (scaled by maximum_thickening)
  gemm_act_kernel<16><<<gGemm, TB, 0, stream>>>(featA, gctx, 2, 256, reg0W, reg0b,
                                                nullptr, nullptr, featB, NF, 1);
  gemm_act_kernel<8><<<gGemm, TB, 0, stream>>>(featB, nullptr, 0, 256, reg1W, reg1b,
                                               nullptr, nullptr, nbbuf, NF, 1);
  head_final_kernel<<<(NF + 7) / 8, TB, 0, stream>>>(nbbuf, reg2W, reg2b, thick,
                                                     out_m, NF);
}